// Net_65025804861672
// MI455X (gfx1250) — compile-verified
//
#include <hip/hip_runtime.h>

#define N_NODES 1000000
#define N_EDGES 16000000
#define N_TILES (N_NODES / 16)   // 62500

typedef __attribute__((ext_vector_type(2)))  float v2f;
typedef __attribute__((ext_vector_type(8)))  float v8f;

// ---------------------------------------------------------------------------
// init: deg starts at 1.0 (self-loop weight), agg1 zeroed, out = b2
// launched with exactly 16*N_NODES threads
// ---------------------------------------------------------------------------
__global__ void k_init(float* __restrict__ deg, float* __restrict__ agg1,
                       float* __restrict__ out, const float* __restrict__ b2) {
    int i = blockIdx.x * blockDim.x + threadIdx.x;
    agg1[i] = 0.0f;
    if (i < N_NODES) {
        deg[i] = 1.0f;              // self-loop contribution to degree
        out[i] = b2[0];
    }
}

// pad W1 (25x16) to W1p (28x16) with zero rows; one block of 448 threads
__global__ void k_padW1(const float* __restrict__ W1, float* __restrict__ W1p) {
    int i = threadIdx.x;            // 0..447
    int k = i >> 4;
    int f = i & 15;
    W1p[i] = (k < 25) ? W1[k * 16 + f] : 0.0f;
}

// ---------------------------------------------------------------------------
// degree accumulation: deg[dst[e]] += w[e]   (exactly N_EDGES threads)
// ---------------------------------------------------------------------------
__global__ void k_deg(const long long* __restrict__ dst,
                      const float* __restrict__ w,
                      float* __restrict__ deg) {
    int e = blockIdx.x * blockDim.x + threadIdx.x;
    atomicAdd(&deg[dst[e]], w[e]);
}

// deg -> dis = rsqrt(deg) in place
__global__ void k_rsqrt(float* __restrict__ dis) {
    int i = blockIdx.x * blockDim.x + threadIdx.x;
    if (i < N_NODES) {
        float d = dis[i];
        dis[i] = (d > 0.0f) ? rsqrtf(d) : 0.0f;
    }
}

// ---------------------------------------------------------------------------
// h1 = x @ W1 via V_WMMA_F32_16X16X4_F32. 8 tiles per 256-thread block,
// one 16x16 output tile per wave. K=25 padded to 28 (7 WMMA steps).
// A reads use a clamped in-bounds index for k>=25; the matching B rows are
// exactly zero (padded), so those products contribute 0. No divergent guards.
// ---------------------------------------------------------------------------
__global__ void __launch_bounds__(256)
k_gemm1(const float* __restrict__ x, const float* __restrict__ W1p,
        float* __restrict__ h1) {
    const int wid  = threadIdx.x >> 5;           // wave within block
    const int lane = threadIdx.x & 31;
    const int tile = blockIdx.x * 8 + wid;
    if (tile >= N_TILES) return;                 // uniform per wave

    const int half = lane >> 4;                  // 0: lanes 0-15, 1: 16-31
    const int l    = lane & 15;
    const long long row = (long long)tile * 16 + l;
    const float* __restrict__ xr = x + row * 25;

    v8f c = {};
    #pragma unroll
    for (int kk = 0; kk < 28; kk += 4) {
        const int k0  = kk + half * 2;           // A VGPR0 / B VGPR0 K index
        const int k1  = k0 + 1;                  // A VGPR1 / B VGPR1 K index
        const int ka0 = (k0 < 25) ? k0 : 24;     // clamp (B row is 0 there)
        const int ka1 = (k1 < 25) ? k1 : 24;
        v2f a, b;
        a[0] = xr[ka0];
        a[1] = xr[ka1];
        b[0] = W1p[k0 * 16 + l];
        b[1] = W1p[k1 * 16 + l];
        c = __builtin_amdgcn_wmma_f32_16x16x4_f32(
                false, a, false, b, (short)0, c, false, false);
    }
    // C/D layout: VGPR r -> M = r + half*8, N = l
    #pragma unroll
    for (int r = 0; r < 8; ++r) {
        const int m = r + half * 8;
        h1[((long long)tile * 16 + m) * 16 + l] = c[r];
    }
}

// ---------------------------------------------------------------------------
// conv1 scatter: agg1[dst] += norm * h1[src], 16 lanes per edge (one per feat)
// index range [0,E) = real edges, [E, E+N) = self loops (src=dst=n, w=1)
// ---------------------------------------------------------------------------
__global__ void k_conv1(const long long* __restrict__ ei,   // [2*E]: src, dst
                        const float* __restrict__ w,
                        const float* __restrict__ dis,
                        const float* __restrict__ h1,
                        float* __restrict__ agg1) {
    const long long t = (long long)blockIdx.x * blockDim.x + threadIdx.x;
    const long long e = t >> 4;
    const int       f = (int)(t & 15);
    long long s, d;
    float nm;
    if (e < N_EDGES) {
        s  = ei[e];
        d  = ei[N_EDGES + e];
        nm = dis[s] * w[e] * dis[d];
    } else {                               // self loop, weight 1
        const long long n = e - N_EDGES;   // grid sized so n < N_NODES
        s = d = n;
        const float ds = dis[n];
        nm = ds * ds;
    }
    atomicAdd(&agg1[d * 16 + f], nm * h1[s * 16 + f]);
}

// ---------------------------------------------------------------------------
// fused: h = elu(agg1 + b1); h2 = h @ W2   (per node, float4-vectorized)
// ---------------------------------------------------------------------------
__global__ void k_elu_h2(const float* __restrict__ agg1,
                         const float* __restrict__ b1,
                         const float* __restrict__ W2,
                         float* __restrict__ h2) {
    const int i = blockIdx.x * blockDim.x + threadIdx.x;
    if (i >= N_NODES) return;
    const float4* __restrict__ a4 = (const float4*)(agg1 + (long long)i * 16);
    const float4* __restrict__ b4 = (const float4*)b1;
    const float4* __restrict__ w4 = (const float4*)W2;
    float acc = 0.0f;
    #pragma unroll
    for (int q = 0; q < 4; ++q) {
        float4 v = a4[q];
        float4 bb = b4[q];
        float4 ww = w4[q];
        float t;
        t = v.x + bb.x; t = (t > 0.0f) ? t : expm1f(t); acc += t * ww.x;
        t = v.y + bb.y; t = (t > 0.0f) ? t : expm1f(t); acc += t * ww.y;
        t = v.z + bb.z; t = (t > 0.0f) ? t : expm1f(t); acc += t * ww.z;
        t = v.w + bb.w; t = (t > 0.0f) ? t : expm1f(t); acc += t * ww.w;
    }
    h2[i] = acc;
}

// ---------------------------------------------------------------------------
// conv2 scatter: out[dst] += norm * h2[src]  (out pre-initialized to b2)
// ---------------------------------------------------------------------------
__global__ void k_conv2(const long long* __restrict__ ei,
                        const float* __restrict__ w,
                        const float* __restrict__ dis,
                        const float* __restrict__ h2,
                        float* __restrict__ out) {
    const long long e = (long long)blockIdx.x * blockDim.x + threadIdx.x;
    long long s, d;
    float nm;
    if (e < N_EDGES) {
        s  = ei[e];
        d  = ei[N_EDGES + e];
        nm = dis[s] * w[e] * dis[d];
    } else if (e < (long long)N_EDGES + N_NODES) {
        const long long n = e - N_EDGES;
        s = d = n;
        const float ds = dis[n];
        nm = ds * ds;
    } else {
        return;
    }
    atomicAdd(&out[d], nm * h2[s]);
}

// ---------------------------------------------------------------------------
extern "C" void kernel_launch(void* const* d_in, const int* in_sizes, int n_in,
                              void* d_out, int out_size, void* d_ws, size_t ws_size,
                              hipStream_t stream) {
    const float*     x  = (const float*)d_in[0];       // [N, 25]
    const long long* ei = (const long long*)d_in[1];   // [2, E] int64
    const float*     w  = (const float*)d_in[2];       // [E]
    const float*     W1 = (const float*)d_in[3];       // [25, 16]
    const float*     b1 = (const float*)d_in[4];       // [16]
    const float*     W2 = (const float*)d_in[5];       // [16, 1]
    const float*     b2 = (const float*)d_in[6];       // [1]
    float*           out = (float*)d_out;              // [N, 1]

    // workspace: 34N floats + 448 = ~136 MB (fits in 192 MB L2 together)
    float* dis  = (float*)d_ws;                        // N      (deg -> rsqrt)
    float* h1   = dis  + N_NODES;                      // 16N    (x @ W1)
    float* agg1 = h1   + 16ll * N_NODES;               // 16N    (conv1 accum)
    float* h2   = agg1 + 16ll * N_NODES;               // N      (elu(h) @ W2)
    float* W1p  = h2   + N_NODES;                      // 448    (padded W1)

    // 1. init deg/agg1/out + pad W1
    k_init<<<(16 * N_NODES) / 256, 256, 0, stream>>>(dis, agg1, out, b2);
    k_padW1<<<1, 448, 0, stream>>>(W1, W1p);
    // 2. degree accumulation + rsqrt
    k_deg<<<N_EDGES / 256, 256, 0, stream>>>(ei + N_EDGES, w, dis);
    k_rsqrt<<<(N_NODES + 255) / 256, 256, 0, stream>>>(dis);
    // 3. h1 = x @ W1 (WMMA, 8 tiles per block, one per wave)
    k_gemm1<<<(N_TILES + 7) / 8, 256, 0, stream>>>(x, W1p, h1);
    // 4. conv1 scatter (16 lanes per edge incl. self loops)
    k_conv1<<<((long long)(N_EDGES + N_NODES) * 16) / 256, 256, 0, stream>>>(
        ei, w, dis, h1, agg1);
    // 5. bias + ELU + (h @ W2) fused
    k_elu_h2<<<(N_NODES + 255) / 256, 256, 0, stream>>>(agg1, b1, W2, h2);
    // 6. conv2 scalar scatter
    k_conv2<<<(N_EDGES + N_NODES + 255) / 256, 256, 0, stream>>>(
        ei, w, dis, h2, out);
}